// YoloLossV2_20315195310297
// MI455X (gfx1250) — compile-verified
//
#include <hip/hip_runtime.h>
#include <hip/hip_bf16.h>
#include <math.h>

// ---------------------------------------------------------------------------
// YOLO-v1 style loss for MI455X (gfx1250, wave32).
//
// Roofline: predicts = 1024*14*14*95*4 B ~= 76 MB read once -> ~3.3 us at
// 23.3 TB/s HBM; refactored FLOPs are negligible => bandwidth bound.
// Strategy: one 256-thread block (8 wave32) per batch; precompute per batch
//   sumPC2 = sum over 588 confidences of p_C^2          (noobject base)
//   S2[cell] = sum_c p_P[cell][c]^2  via V_WMMA_F32_16X16X4_F32 with B=ones
// then each object only needs a small rectangle sum + CIoU at 3 center boxes.
// The S2 pass batches all 20 row loads (global_load_b64 x20, single wait)
// before the 20-deep WMMA chain; bounds handling is a branch-free address
// clamp + select so no EXEC masking breaks load clustering.
// ---------------------------------------------------------------------------

typedef float v2f __attribute__((ext_vector_type(2)));
typedef float v8f __attribute__((ext_vector_type(8)));

#define CELLS   14
#define NCLS    80
#define NBPC    3
#define CHAN    95            // NCLS + 5*NBPC
#define NCELL2  196           // 14*14
#define PRED_STRIDE (NCELL2*CHAN) // 18620 floats per batch
#define CSF     32.0f         // IMG / CELL
#define IMGF    448.0f
#define EPSF    1e-9f
#define FOUR_OVER_PI2 0.40528473456f

__device__ __forceinline__ float waveSum(float v) {
  #pragma unroll
  for (int off = 16; off > 0; off >>= 1) v += __shfl_xor(v, off, 32);
  return v;
}

__global__ void __launch_bounds__(256)
yolo_loss_kernel(const float* __restrict__ pred,
                 const float* __restrict__ labels,
                 const int*   __restrict__ objn,
                 float*       __restrict__ ws,
                 int M) {
  __shared__ float S2[208];        // per-cell sum of squared class scores (13 tiles of 16)
  __shared__ float lab[150];       // this batch's labels (M*5 <= 150)
  __shared__ float pc2Slots[8];    // per-wave partial of sum(p_C^2)
  __shared__ float lossSlots[8];   // per-wave loss accumulators
  __shared__ float sPC2sh;

  const int b    = blockIdx.x;
  const int tid  = threadIdx.x;
  const int lane = tid & 31;
  const int wave = tid >> 5;
  const float* pb = pred + (size_t)b * PRED_STRIDE;

  // Prefetch the batch slice into the cache hierarchy (global_prefetch_b8).
  for (int off = tid * 32; off < PRED_STRIDE; off += 256 * 32)
    __builtin_prefetch(pb + off, 0, 0);

  // Stage labels in LDS.
  int lcount = M * 5; if (lcount > 150) lcount = 150;
  if (tid < lcount) lab[tid] = labels[(size_t)b * (M * 5) + tid];

  // ---- per-batch sum of p_C^2 over all 196 cells x 3 boxes ----
  float pc2 = 0.0f;
  if (tid < NCELL2) {
    const float* c = pb + tid * CHAN + NCLS;
    float c0 = c[0], c1 = c[1], c2 = c[2];
    pc2 = c0*c0 + c1*c1 + c2*c2;
  }
  pc2 = waveSum(pc2);
  if (lane == 0) pc2Slots[wave] = pc2;

  // ---- S2[cell] = sum_c p_P^2 via f32 WMMA (A = squared scores, B = ones) ----
  // A layout (16x4 f32): lanes 0-15 hold M=0..15 with K=0,1 in v[0],v[1];
  // lanes 16-31 hold the same rows with K=2,3.
  const int rsub = lane & 15;
  const int koff = (lane >> 4) * 2;
  v2f bOnes; bOnes.x = 1.0f; bOnes.y = 1.0f;
  for (int tile = wave; tile < 13; tile += 8) {       // uniform per wave: EXEC all-1s at WMMA
    int row = tile * 16 + rsub;
    bool ok = (row < NCELL2);
    int rowc = ok ? row : (NCELL2 - 1);               // branch-free clamp: loads stay in-bounds
    const float* rp = pb + rowc * CHAN + koff;

    // Batch all 20 row loads first (merged to global_load_b64, single wait).
    float a0v[20], a1v[20];
    #pragma unroll
    for (int kk = 0; kk < 20; ++kk) {
      a0v[kk] = rp[kk * 4];
      a1v[kk] = rp[kk * 4 + 1];
    }

    v8f acc = {};
    #pragma unroll
    for (int kk = 0; kk < 20; ++kk) {
      float a0 = ok ? a0v[kk] : 0.0f;                 // v_cndmask, no EXEC games
      float a1 = ok ? a1v[kk] : 0.0f;
      v2f a; a.x = a0 * a0; a.y = a1 * a1;
      acc = __builtin_amdgcn_wmma_f32_16x16x4_f32(false, a, false, bOnes,
                                                  (short)0, acc, false, false);
    }
    // D column 0: VGPR r / lane 0 -> M=r ; VGPR r / lane 16 -> M=8+r.
    if (lane == 0) {
      #pragma unroll
      for (int r = 0; r < 8; ++r) S2[tile * 16 + r] = acc[r];
    } else if (lane == 16) {
      #pragma unroll
      for (int r = 0; r < 8; ++r) S2[tile * 16 + 8 + r] = acc[r];
    }
  }
  __syncthreads();
  if (tid == 0) {                       // fixed-order sum -> deterministic
    float s = 0.0f;
    #pragma unroll
    for (int i = 0; i < 8; ++i) s += pc2Slots[i];
    sPC2sh = s;
  }
  __syncthreads();
  const float sPC2 = sPC2sh;

  // ---- per-object loss; waves take objects round-robin ----
  int nobj = objn[b];
  if (nobj > M) nobj = M;
  if (nobj < 0) nobj = 0;
  const float inv = 1.0f / CSF;

  float waveAcc = 0.0f;
  for (int m = wave; m < nobj; m += 8) {
    float x = lab[m*5+0], y = lab[m*5+1], w = lab[m*5+2], h = lab[m*5+3];
    int cls = (int)lab[m*5+4];
    if (cls < 0) cls = 0; if (cls >= NCLS) cls = NCLS - 1;

    int ix0 = (int)floorf((x - 0.5f*w) * inv);
    int ix1 = (int)fminf(ceilf((x + 0.5f*w) * inv), (float)CELLS);
    int iy0 = (int)floorf((y - 0.5f*h) * inv);
    int iy1 = (int)fminf(ceilf((y + 0.5f*h) * inv), (float)CELLS);
    if (ix0 < 0) ix0 = 0; if (iy0 < 0) iy0 = 0;
    int nx = ix1 - ix0; if (nx < 0) nx = 0;
    int ny = iy1 - iy0; if (ny < 0) ny = 0;
    int ncells = nx * ny;
    int cx = (int)floorf(x * inv);
    int cy = (int)floorf(y * inv);

    // class loss over object rectangle: sum_c (p_P - onehot)^2 = S2 - 2*p_P[cls] + 1
    float classPart = 0.0f;
    for (int idx = lane; idx < ncells; idx += 32) {
      int r = iy0 + idx / nx;
      int c = ix0 + idx % nx;
      int cell = r * CELLS + c;
      float pp = pb[cell * CHAN + cls];
      classPart += S2[cell] - 2.0f * pp + 1.0f;
    }

    // CIoU of the 3 boxes at the center cell (lanes 0..2; others duplicate box 2)
    int bidx = lane < 3 ? lane : 2;
    int cbase = (cy * CELLS + cx) * CHAN;
    float pc = pb[cbase + NCLS + bidx];
    const float* t = pb + cbase + NCLS + NBPC + bidx * 4;
    float bx = t[0] * CSF + (float)cx * CSF;
    float by = t[1] * CSF + (float)cy * CSF;
    float bw = t[2] * IMGF;
    float bh = t[3] * IMGF;

    float b1x1 = bx - 0.5f*bw, b1x2 = bx + 0.5f*bw;
    float b1y1 = by - 0.5f*bh, b1y2 = by + 0.5f*bh;
    float b2x1 = x  - 0.5f*w,  b2x2 = x  + 0.5f*w;
    float b2y1 = y  - 0.5f*h,  b2y2 = y  + 0.5f*h;
    float iw = fmaxf(fminf(b1x2, b2x2) - fmaxf(b1x1, b2x1), 0.0f);
    float ih = fmaxf(fminf(b1y2, b2y2) - fmaxf(b1y1, b2y1), 0.0f);
    float inter = iw * ih;
    float uni = bw*bh + w*h - inter;
    float iou = inter / (uni + EPSF);
    float cd = (bx - x)*(bx - x) + (by - y)*(by - y);
    // reference "enclose" box is computed from (cx,cy,w,h) fields directly:
    float el = fminf(bx, x), er = fmaxf(bw, w);
    float et = fminf(by, y), eb = fmaxf(bh, h);
    float ed = (er - el)*(er - el) + (eb - et)*(eb - et);
    float dv = atanf(w / (h + EPSF)) - atanf(bw / (bh + EPSF));
    float v  = FOUR_OVER_PI2 * dv * dv;
    float alpha = v / (1.0f - iou + v + EPSF);
    float ci = iou - cd / (ed + EPSF) - alpha * v;

    float c0 = __shfl(ci, 0, 32);
    float c1 = __shfl(ci, 1, 32);
    float c2 = __shfl(ci, 2, 32);
    float maxci = fmaxf(fmaxf(c0, c1), c2);

    float part = 0.5f * classPart;                // CLASS_SCALE = 1
    if (lane < 3 && ci >= maxci) {                // responsible box(es), ties as in ref
      float d = pc - ci;
      part += 0.5f * d * d;                       // OBJECT_SCALE = 1
      part -= 0.25f * pc * pc;                    // noobject: remove I spots
      float dx = (bx - x) * inv;
      float dy = (by - y) * inv;
      float dsw = sqrtf(fminf(fmaxf(bw, 0.0f), IMGF)) - sqrtf(fabsf(w));
      float dsh = sqrtf(fminf(fmaxf(bh, 0.0f), IMGF)) - sqrtf(fabsf(h));
      part += 2.5f * (dx*dx + dy*dy + (dsw*dsw + dsh*dsh) * (1.0f/IMGF));
    }
    if (lane == 0) part += 0.25f * sPC2;          // 0.5 * NOOBJECT_SCALE * sum(p_C^2)
    part = waveSum(part);
    if (lane == 0) waveAcc += part;
  }

  if (lane == 0) lossSlots[wave] = waveAcc;       // every wave writes (0 if no objects)
  __syncthreads();
  if (tid == 0) {
    float s = 0.0f;
    #pragma unroll
    for (int i = 0; i < 8; ++i) s += lossSlots[i];
    ws[b] = s;
  }
}

__global__ void __launch_bounds__(256)
final_reduce_kernel(const float* __restrict__ ws, float* __restrict__ out, int B) {
  __shared__ float s[256];
  int t = threadIdx.x;
  float v = 0.0f;
  for (int i = t; i < B; i += 256) v += ws[i];   // fixed-order per-thread sum
  s[t] = v;
  __syncthreads();
  for (int off = 128; off > 0; off >>= 1) {
    if (t < off) s[t] += s[t + off];
    __syncthreads();
  }
  if (t == 0) out[0] = s[0] / (float)B;
}

extern "C" void kernel_launch(void* const* d_in, const int* in_sizes, int n_in,
                              void* d_out, int out_size, void* d_ws, size_t ws_size,
                              hipStream_t stream) {
  const float* pred   = (const float*)d_in[0];
  const float* labels = (const float*)d_in[1];
  const int*   objnum = (const int*)d_in[2];
  int B = in_sizes[2];                 // objects_num has one entry per batch
  int M = in_sizes[1] / (B * 5);       // labels are (B, M, 5)
  float* ws = (float*)d_ws;            // per-batch losses (B floats)

  yolo_loss_kernel<<<B, 256, 0, stream>>>(pred, labels, objnum, ws, M);
  final_reduce_kernel<<<1, 256, 0, stream>>>(ws, (float*)d_out, B);
}